// S4Kernel_35055523070041
// MI455X (gfx1250) — compile-verified
//
#include <hip/hip_runtime.h>
#include <hip/hip_bf16.h>
#include <math.h>

// ---------------------------------------------------------------------------
// S4 generating-function kernel for MI455X (gfx1250, wave32).
//   N = 512 modes, L = 65536 output taps.
//   Stage A: terms[n][4] from Vc@B matvec (fused, trivial).
//   Stage B: atRoots[l] = c_l * (k00 - k01*k10/(1+k11)),
//            k_j = sum_n terms[n][j] / (i*G_l - Lambda_n),  G_l=(2/step)tan(pi l/L)
//            -> dominant VALU work (~1.4 GFLOP), LDS-resident Lambda/terms.
//   Stage C: length-65536 IFFT as four-step 256x256 decomposition; each
//            256-pt DFT batch = 256x256x256 f32 GEMM on V_WMMA_F32_16X16X4_F32.
// ---------------------------------------------------------------------------

typedef float v2f __attribute__((ext_vector_type(2)));
typedef float v8f __attribute__((ext_vector_type(8)));

#define S4_N   512
#define S4_L   65536
#define PI_F   3.14159265358979323846f

__device__ __forceinline__ v8f wmma_f32(v2f a, v2f b, v8f c) {
    // 8 args: (neg_a, A, neg_b, B, c_mod, C, reuse_a, reuse_b)
    return __builtin_amdgcn_wmma_f32_16x16x4_f32(false, a, false, b,
                                                 (short)0, c, false, false);
}

// ---------------------------------------------------------------------------
// Stage A: Bc[n] = (Vc @ B)[n]; terms[n][0..3] = {a0*b0, a0*b1, a1*b0, a1*b1}
//   a0 = conj(Ct), a1 = conj(q), b0 = Bc, b1 = p.  Thread n owns row n.
// ---------------------------------------------------------------------------
__global__ __launch_bounds__(256) void prep_terms_kernel(
    const float* __restrict__ Vc_re, const float* __restrict__ Vc_im,
    const float* __restrict__ B,
    const float* __restrict__ p_re, const float* __restrict__ p_im,
    const float* __restrict__ q_re, const float* __restrict__ q_im,
    const float* __restrict__ Ct,
    float* __restrict__ terms)
{
    __shared__ float sB[S4_N];
    const int tid = threadIdx.x;
    for (int i = tid; i < S4_N; i += blockDim.x) sB[i] = B[i];
    __syncthreads();

    const int n = blockIdx.x * blockDim.x + tid;
    if (n >= S4_N) return;

    const float* vr = Vc_re + (size_t)n * S4_N;
    const float* vi = Vc_im + (size_t)n * S4_N;
    float br = 0.f, bi = 0.f;
    #pragma unroll 8
    for (int k = 0; k < S4_N; ++k) {
        br = fmaf(vr[k], sB[k], br);
        bi = fmaf(vi[k], sB[k], bi);
    }

    const float a0r = Ct[2 * n],     a0i = -Ct[2 * n + 1];
    const float a1r = q_re[n],       a1i = -q_im[n];
    const float b1r = p_re[n],       b1i =  p_im[n];

    float* T = terms + n * 8;
    T[0] = a0r * br  - a0i * bi;   T[1] = a0r * bi  + a0i * br;   // a0*b0
    T[2] = a0r * b1r - a0i * b1i;  T[3] = a0r * b1i + a0i * b1r;  // a0*b1
    T[4] = a1r * br  - a1i * bi;   T[5] = a1r * bi  + a1i * br;   // a1*b0
    T[6] = a1r * b1r - a1i * b1i;  T[7] = a1r * b1i + a1i * b1r;  // a1*b1
}

// ---------------------------------------------------------------------------
// 256-point inverse-DFT matrix: F[a][b] = exp(+2*pi*i*a*b/256)
// ---------------------------------------------------------------------------
__global__ __launch_bounds__(256) void fmat_kernel(float* __restrict__ Fr,
                                                   float* __restrict__ Fi)
{
    const int t = blockIdx.x * blockDim.x + threadIdx.x;  // 65536 threads
    const int a = t >> 8, b = t & 255;
    const int ab = (a * b) & 255;
    float s, c;
    __sincosf((float)ab * (2.0f * PI_F / 256.0f), &s, &c);
    Fr[t] = c;
    Fi[t] = s;
}

// ---------------------------------------------------------------------------
// Stage B: atRoots.  g_l = i*G, G = (2/step)*tan(pi*l/L); c_l = 1 + i*tan(pi*l/L)
// r = 1/(g - Lambda) = (-Lr - i*(G - Li)) / (Lr^2 + (G - Li)^2)
// ---------------------------------------------------------------------------
__global__ __launch_bounds__(256) void atroots_kernel(
    const float* __restrict__ Lambda_re, const float* __restrict__ Lambda_im,
    const float* __restrict__ terms, const float* __restrict__ log_step,
    float* __restrict__ Ar, float* __restrict__ Ai)
{
    __shared__ float sLr[S4_N];
    __shared__ float sLi[S4_N];
    __shared__ float sT[S4_N * 8];

    const int tid = threadIdx.x;
    for (int i = tid; i < S4_N; i += 256) { sLr[i] = Lambda_re[i]; sLi[i] = Lambda_im[i]; }
    for (int i = tid; i < S4_N * 8; i += 256) sT[i] = terms[i];
    __syncthreads();

    const int l = blockIdx.x * 256 + tid;
    const float two_over_step = 2.0f * expf(-log_step[0]);
    const float t = tanf((float)l * (PI_F / (float)S4_L));
    const float G = two_over_step * t;

    float k00r = 0.f, k00i = 0.f, k01r = 0.f, k01i = 0.f;
    float k10r = 0.f, k10i = 0.f, k11r = 0.f, k11i = 0.f;

    #pragma unroll 4
    for (int n = 0; n < S4_N; ++n) {
        const float dr = -sLr[n];
        const float di = G - sLi[n];
        const float inv = 1.0f / fmaf(dr, dr, di * di);
        const float rr = dr * inv;
        const float ri = -di * inv;
        const float* T = &sT[n * 8];
        k00r = fmaf(rr, T[0], fmaf(-ri, T[1], k00r));
        k00i = fmaf(rr, T[1], fmaf( ri, T[0], k00i));
        k01r = fmaf(rr, T[2], fmaf(-ri, T[3], k01r));
        k01i = fmaf(rr, T[3], fmaf( ri, T[2], k01i));
        k10r = fmaf(rr, T[4], fmaf(-ri, T[5], k10r));
        k10i = fmaf(rr, T[5], fmaf( ri, T[4], k10i));
        k11r = fmaf(rr, T[6], fmaf(-ri, T[7], k11r));
        k11i = fmaf(rr, T[7], fmaf( ri, T[6], k11i));
    }

    // w = k01*k10 / (1 + k11)
    const float der = 1.0f + k11r, dei = k11i;
    const float dinv = 1.0f / fmaf(der, der, dei * dei);
    const float numr = k01r * k10r - k01i * k10i;
    const float numi = k01r * k10i + k01i * k10r;
    const float wr = (numr * der + numi * dei) * dinv;
    const float wi = (numi * der - numr * dei) * dinv;
    const float sr = k00r - wr, si = k00i - wi;
    // at = (1 + i*t) * s
    Ar[l] = sr - t * si;
    Ai[l] = si + t * sr;
}

// ---------------------------------------------------------------------------
// Twiddle + transpose: Z[k1][n1] = Y[n1][k1] * exp(+2*pi*i*k1*n1/65536)
// Consecutive threads write consecutive Z addresses (coalesced stores).
// ---------------------------------------------------------------------------
__global__ __launch_bounds__(256) void twiddle_kernel(
    const float* __restrict__ Yr, const float* __restrict__ Yi,
    float* __restrict__ Zr, float* __restrict__ Zi)
{
    const int tidx = blockIdx.x * blockDim.x + threadIdx.x;  // 65536
    const int n1 = tidx & 255, k1 = tidx >> 8;
    const float yr = Yr[n1 * 256 + k1];
    const float yi = Yi[n1 * 256 + k1];
    float s, c;
    __sincosf((float)(k1 * n1) * (2.0f * PI_F / (float)S4_L), &s, &c);
    Zr[tidx] = yr * c - yi * s;
    Zi[tidx] = yr * s + yi * c;
}

// ---------------------------------------------------------------------------
// Complex 256x256x256 GEMM on f32 WMMA:  Y = F * X  (row-major, ld=256).
// One wave per 16x16 output tile; 256 tiles -> 32 blocks x 8 waves.
// A 16x4 f32 layout (ISA 7.12.2): lane t -> row t&15, K = (t>>4)*2 + {0,1}.
// B 4x16 mirrored: lane t -> col t&15, K = (t>>4)*2 + {0,1}.
// C/D: VGPR v -> row v + (t>>4)*8, col t&15.
// ---------------------------------------------------------------------------
__global__ __launch_bounds__(256) void cgemm256_kernel(
    const float* __restrict__ Fr, const float* __restrict__ Fi,
    const float* __restrict__ Xr, const float* __restrict__ Xi,
    float* __restrict__ Yr, float* __restrict__ Yi)
{
    const int lane  = threadIdx.x & 31;
    const int wave  = blockIdx.x * (blockDim.x >> 5) + (threadIdx.x >> 5);
    const int tm    = wave >> 4, tn = wave & 15;
    const int mrow  = tm * 16 + (lane & 15);
    const int ncol  = tn * 16 + (lane & 15);
    const int kb    = (lane >> 4) << 1;           // 0 or 2

    v8f c_rr = {}, c_ii = {}, c_im = {};

    for (int kk = 0; kk < 256; kk += 4) {
        const v2f ar = *(const v2f*)(Fr + mrow * 256 + kk + kb);
        const v2f ai = *(const v2f*)(Fi + mrow * 256 + kk + kb);
        v2f br, bi;
        br.x = Xr[(kk + kb) * 256 + ncol];  br.y = Xr[(kk + kb + 1) * 256 + ncol];
        bi.x = Xi[(kk + kb) * 256 + ncol];  bi.y = Xi[(kk + kb + 1) * 256 + ncol];
        c_rr = wmma_f32(ar, br, c_rr);   // sum Fr*Xr
        c_ii = wmma_f32(ai, bi, c_ii);   // sum Fi*Xi
        c_im = wmma_f32(ar, bi, c_im);   // sum Fr*Xi
        c_im = wmma_f32(ai, br, c_im);   //  + Fi*Xr
    }

    const int rbase = tm * 16 + ((lane >> 4) << 3);
    #pragma unroll
    for (int v = 0; v < 8; ++v) {
        Yr[(rbase + v) * 256 + ncol] = c_rr[v] - c_ii[v];
        Yi[(rbase + v) * 256 + ncol] = c_im[v];
    }
}

// ---------------------------------------------------------------------------
// Real-part-only GEMM: out = Re(F * Z) / L = (Fr*Zr - Fi*Zi) / 65536.
// out[n1 + 256*n2] = Out[n2][n1]  ->  row-major store straight into d_out.
// ---------------------------------------------------------------------------
__global__ __launch_bounds__(256) void rgemm256_kernel(
    const float* __restrict__ Fr, const float* __restrict__ Fi,
    const float* __restrict__ Zr, const float* __restrict__ Zi,
    float* __restrict__ out)
{
    const int lane  = threadIdx.x & 31;
    const int wave  = blockIdx.x * (blockDim.x >> 5) + (threadIdx.x >> 5);
    const int tm    = wave >> 4, tn = wave & 15;
    const int mrow  = tm * 16 + (lane & 15);
    const int ncol  = tn * 16 + (lane & 15);
    const int kb    = (lane >> 4) << 1;

    v8f c_rr = {}, c_ii = {};

    for (int kk = 0; kk < 256; kk += 4) {
        const v2f ar = *(const v2f*)(Fr + mrow * 256 + kk + kb);
        const v2f ai = *(const v2f*)(Fi + mrow * 256 + kk + kb);
        v2f br, bi;
        br.x = Zr[(kk + kb) * 256 + ncol];  br.y = Zr[(kk + kb + 1) * 256 + ncol];
        bi.x = Zi[(kk + kb) * 256 + ncol];  bi.y = Zi[(kk + kb + 1) * 256 + ncol];
        c_rr = wmma_f32(ar, br, c_rr);
        c_ii = wmma_f32(ai, bi, c_ii);
    }

    const float scale = 1.0f / (float)S4_L;
    const int rbase = tm * 16 + ((lane >> 4) << 3);
    #pragma unroll
    for (int v = 0; v < 8; ++v) {
        out[(rbase + v) * 256 + ncol] = (c_rr[v] - c_ii[v]) * scale;
    }
}

// ---------------------------------------------------------------------------
// Launch. Inputs (setup_inputs order):
//  0 Lambda_re(512) 1 Lambda_im(512) 2 p_re 3 p_im 4 q_re 5 q_im
//  6 Vc_re(512*512) 7 Vc_im 8 Ct(1024) 9 B(512) 10 log_step(1) 11 L(1)
// Workspace (floats): terms 4096 | Fr 65536 | Fi 65536 | Ar 65536 | Ai 65536 |
//                     Yr 65536 | Yi 65536   (~1.59 MB total; Z reuses Ar/Ai)
// ---------------------------------------------------------------------------
extern "C" void kernel_launch(void* const* d_in, const int* in_sizes, int n_in,
                              void* d_out, int out_size, void* d_ws, size_t ws_size,
                              hipStream_t stream)
{
    const float* Lambda_re = (const float*)d_in[0];
    const float* Lambda_im = (const float*)d_in[1];
    const float* p_re      = (const float*)d_in[2];
    const float* p_im      = (const float*)d_in[3];
    const float* q_re      = (const float*)d_in[4];
    const float* q_im      = (const float*)d_in[5];
    const float* Vc_re     = (const float*)d_in[6];
    const float* Vc_im     = (const float*)d_in[7];
    const float* Ct        = (const float*)d_in[8];
    const float* B         = (const float*)d_in[9];
    const float* log_step  = (const float*)d_in[10];
    float*       out       = (float*)d_out;

    float* w     = (float*)d_ws;
    float* terms = w;                 // 4096
    float* Fr    = w + 4096;          // 65536
    float* Fi    = w + 69632;         // 65536
    float* Ar    = w + 135168;        // 65536 (X_re, later Z_re)
    float* Ai    = w + 200704;        // 65536 (X_im, later Z_im)
    float* Yr    = w + 266240;        // 65536
    float* Yi    = w + 331776;        // 65536

    prep_terms_kernel<<<2, 256, 0, stream>>>(Vc_re, Vc_im, B, p_re, p_im,
                                             q_re, q_im, Ct, terms);
    fmat_kernel<<<256, 256, 0, stream>>>(Fr, Fi);
    atroots_kernel<<<256, 256, 0, stream>>>(Lambda_re, Lambda_im, terms,
                                            log_step, Ar, Ai);
    cgemm256_kernel<<<32, 256, 0, stream>>>(Fr, Fi, Ar, Ai, Yr, Yi);
    twiddle_kernel<<<256, 256, 0, stream>>>(Yr, Yi, /*Z=*/Ar, Ai);
    rgemm256_kernel<<<32, 256, 0, stream>>>(Fr, Fi, Ar, Ai, out);
}